// FullAttention_70196945486336
// MI455X (gfx1250) — compile-verified
//
#include <hip/hip_runtime.h>

#define BB 4
#define HH 8
#define LL 1024
#define SS 1024
#define EE 64
#define GG 4   // HH / HEAD_KS

typedef __attribute__((ext_vector_type(2))) float v2f;
typedef __attribute__((ext_vector_type(4))) float v4f;
typedef __attribute__((ext_vector_type(8))) float v8f;

// ---------------------------------------------------------------------------
// K1: scores = scale * Q K^T, causal-masked to 0, via V_WMMA_F32_16X16X4_F32.
// Q tile (shared by all 8 waves of the block) staged in LDS; A/B fragments are
// contiguous 8-byte pairs ({K0,K1} lanes 0-15 / {K2,K3} lanes 16-31).
// ---------------------------------------------------------------------------
__global__ __launch_bounds__(256) void qk_wmma(const float* __restrict__ q,
                                               const float* __restrict__ kk,
                                               float* __restrict__ s_out) {
  __shared__ float sq[16 * 68];            // 16 rows x 64, padded stride 68
  const int tid = threadIdx.x;
  const int li = blockIdx.x;               // query tile
  const int bh = blockIdx.z;
  const int b = bh >> 3, h = bh & 7;

  // Cooperative stage of Q tile [16 x 64] (one b128 per thread).
  {
    const int row = tid >> 4;              // 0..15
    const int col = (tid & 15) * 4;        // 0..60
    const v4f t = *(const v4f*)&q[(((size_t)b * LL + li * 16 + row) * HH + h) * EE + col];
    sq[row * 68 + col + 0] = t[0];
    sq[row * 68 + col + 1] = t[1];
    sq[row * 68 + col + 2] = t[2];
    sq[row * 68 + col + 3] = t[3];
  }
  __syncthreads();

  const int wave = tid >> 5;
  const int lane = tid & 31;
  const int si = blockIdx.y * 8 + wave;    // key tile
  if (si > li) return;                     // strictly-upper tiles never read

  const int hi = lane >> 4;                // lane-half: K-pair select
  const int n  = lane & 15;
  const int ncol = si * 16 + n;

  v8f c = {};
#pragma unroll
  for (int k0 = 0; k0 < EE; k0 += 4) {
    const v2f a  = *(const v2f*)&sq[n * 68 + k0 + 2 * hi];              // ds_load_b64
    const v2f bv = *(const v2f*)&kk[(((size_t)b * LL + ncol) * HH + h) * EE + k0 + 2 * hi];
    c = __builtin_amdgcn_wmma_f32_16x16x4_f32(false, a, false, bv, (short)0, c,
                                              false, false);
  }
  const float scale = 0.125f;  // 1/sqrt(64)
#pragma unroll
  for (int r = 0; r < 8; ++r) {
    const int row = li * 16 + r + hi * 8;  // C/D layout: VGPR r -> M=r / M=r+8
    const int col = si * 16 + n;
    const float v = (col <= row) ? c[r] * scale : 0.f;  // keep-mask -> 0
    s_out[(((size_t)bh) * LL + row) * SS + col] = v;
  }
}

// ---------------------------------------------------------------------------
// wave32 butterfly reductions (xor shuffle), then one LDS exchange per block.
// ---------------------------------------------------------------------------
__device__ __forceinline__ float waveReduceMax(float v) {
#pragma unroll
  for (int off = 16; off; off >>= 1) v = fmaxf(v, __shfl_xor(v, off, 32));
  return v;
}
__device__ __forceinline__ float waveReduceSum(float v) {
#pragma unroll
  for (int off = 16; off; off >>= 1) v += __shfl_xor(v, off, 32);
  return v;
}

// ---------------------------------------------------------------------------
// K2: 3x3 causal conv (all 8 heads) + wpsm head mix + causal softmax.
// One block per (b,l); thread t owns columns [4t, 4t+3] (6-wide tap window
// loaded once per head-row); mixed rows for 8 heads live in LDS (32 KB).
// ---------------------------------------------------------------------------
__global__ __launch_bounds__(256) void mix_softmax(const float* __restrict__ sc,
                                                   const float* __restrict__ mta_k,
                                                   const float* __restrict__ wpsm,
                                                   float* __restrict__ Aout) {
  __shared__ float smix[HH][SS];
  __shared__ float red[8];                 // one slot per wave
  __shared__ float skk[HH * 9];
  __shared__ float swp[HH * HH];
  const int tid = threadIdx.x;
  const int wave = tid >> 5, lane = tid & 31;
  const int l = blockIdx.x, b = blockIdx.y;
  if (tid < HH * 9) skk[tid] = mta_k[tid];
  if (tid < HH * HH) swp[tid] = wpsm[tid];
  __syncthreads();

  const float NEG_INF = -__builtin_inff();
  const int s0 = tid * 4;                  // 256 threads x 4 cols == 1024

  __builtin_prefetch(&sc[(((size_t)(b * HH)) * LL + l) * SS + s0], 0, 1);

  float conv[HH][4];
#pragma unroll
  for (int h = 0; h < HH; ++h) {
    float acc0 = 0.f, acc1 = 0.f, acc2 = 0.f, acc3 = 0.f;
#pragma unroll
    for (int i = 0; i < 3; ++i) {
      const int lr = l + i - 2;            // query top-pad = kq-1 = 2
      if (lr < 0) continue;
      const size_t rb = (((size_t)(b * HH + h)) * LL + lr) * SS;
      float vals[6];
#pragma unroll
      for (int t = 0; t < 6; ++t) {
        const int cc = s0 - 1 + t;         // symmetric key pad = 1
        vals[t] = (cc >= 0 && cc <= lr) ? sc[rb + cc] : 0.f;  // pad/keep -> 0
      }
#pragma unroll
      for (int j = 0; j < 3; ++j) {
        const float w = skk[h * 9 + i * 3 + j];
        acc0 += w * vals[0 + j];
        acc1 += w * vals[1 + j];
        acc2 += w * vals[2 + j];
        acc3 += w * vals[3 + j];
      }
    }
    conv[h][0] = acc0; conv[h][1] = acc1; conv[h][2] = acc2; conv[h][3] = acc3;
  }
#pragma unroll
  for (int g = 0; g < HH; ++g) {
#pragma unroll
    for (int u = 0; u < 4; ++u) {
      float m = 0.f;
#pragma unroll
      for (int h = 0; h < HH; ++h) m += swp[g * HH + h] * conv[h][u];
      smix[g][s0 + u] = (s0 + u <= l) ? m : NEG_INF;   // causal -inf
    }
  }
  __syncthreads();

  for (int g = 0; g < HH; ++g) {
    // block max: wave32 butterfly + 8-slot LDS combine
    float lm = NEG_INF;
#pragma unroll
    for (int u = 0; u < 4; ++u) lm = fmaxf(lm, smix[g][s0 + u]);
    lm = waveReduceMax(lm);
    if (lane == 0) red[wave] = lm;
    __syncthreads();
    float gm = red[0];
#pragma unroll
    for (int w = 1; w < 8; ++w) gm = fmaxf(gm, red[w]);
    __syncthreads();

    float ls = 0.f;
#pragma unroll
    for (int u = 0; u < 4; ++u) ls += __expf(smix[g][s0 + u] - gm);
    ls = waveReduceSum(ls);
    if (lane == 0) red[wave] = ls;
    __syncthreads();
    float gs = red[0];
#pragma unroll
    for (int w = 1; w < 8; ++w) gs += red[w];
    __syncthreads();
    const float inv = 1.f / gs;

    const size_t ob = (((size_t)(b * HH + g)) * LL + l) * SS;
    v4f o;
#pragma unroll
    for (int u = 0; u < 4; ++u) o[u] = __expf(smix[g][s0 + u] - gm) * inv;
    *(v4f*)&Aout[ob + s0] = o;             // exp(-inf)=0 beyond diagonal
  }
}

// ---------------------------------------------------------------------------
// K3: post-softmax 3x3 conv + keep-mask + 2-head group mixing.
// One block per (b, group g, l); thread t owns columns [4t, 4t+3].
// ---------------------------------------------------------------------------
__global__ __launch_bounds__(256) void conv2_headmix(const float* __restrict__ Ain,
                                                     const float* __restrict__ mta_k2,
                                                     const float* __restrict__ head_k,
                                                     float* __restrict__ Aout) {
  const int tid = threadIdx.x;
  const int l = blockIdx.x, g = blockIdx.y, b = blockIdx.z;
  __shared__ float skk[2 * 9];
  __shared__ float shk[4];
  if (tid < 18) skk[tid] = mta_k2[(2 * g) * 9 + tid];
  if (tid < 4) shk[tid] = head_k[g * 4 + tid];
  __syncthreads();

  const int s0 = tid * 4;
  float vj[2][4];
#pragma unroll
  for (int jj = 0; jj < 2; ++jj) {
    const int h = 2 * g + jj;
    float acc0 = 0.f, acc1 = 0.f, acc2 = 0.f, acc3 = 0.f;
#pragma unroll
    for (int i = 0; i < 3; ++i) {
      const int lr = l + i - 2;
      if (lr < 0) continue;
      const size_t rb = (((size_t)(b * HH + h)) * LL + lr) * SS;
      float vals[6];
#pragma unroll
      for (int t = 0; t < 6; ++t) {
        const int cc = s0 - 1 + t;
        vals[t] = (cc >= 0 && cc < SS) ? Ain[rb + cc] : 0.f;  // A already 0 masked
      }
#pragma unroll
      for (int j = 0; j < 3; ++j) {
        const float w = skk[jj * 9 + i * 3 + j];
        acc0 += w * vals[0 + j];
        acc1 += w * vals[1 + j];
        acc2 += w * vals[2 + j];
        acc3 += w * vals[3 + j];
      }
    }
    vj[jj][0] = acc0; vj[jj][1] = acc1; vj[jj][2] = acc2; vj[jj][3] = acc3;
  }
  // keep-mask after conv, then head-group mix: out_k = sum_j A_j * hk[g,j,k]
  v4f o0, o1;
#pragma unroll
  for (int u = 0; u < 4; ++u) {
    const float a0 = (s0 + u <= l) ? vj[0][u] : 0.f;
    const float a1 = (s0 + u <= l) ? vj[1][u] : 0.f;
    o0[u] = a0 * shk[0] + a1 * shk[2];
    o1[u] = a0 * shk[1] + a1 * shk[3];
  }
  *(v4f*)&Aout[(((size_t)(b * HH + 2 * g + 0)) * LL + l) * SS + s0] = o0;
  *(v4f*)&Aout[(((size_t)(b * HH + 2 * g + 1)) * LL + l) * SS + s0] = o1;
}

// ---------------------------------------------------------------------------
// K4: out = A V via V_WMMA_F32_16X16X4_F32. The V panel (b,h,et) is shared by
// all 8 waves of the block -> staged once in LDS (transposed, padded stride).
// kend is always a multiple of 16 -> outer loop steps 16 with 4 unrolled WMMAs
// per iteration (guaranteed 4x v_wmma body, batched ds/global loads).
// ---------------------------------------------------------------------------
__global__ __launch_bounds__(256) void av_wmma(const float* __restrict__ A,
                                               const float* __restrict__ vv,
                                               float* __restrict__ out) {
  __shared__ float sv[16 * 1028];          // svT[e][s], padded stride 1028
  const int tid = threadIdx.x;
  const int et = blockIdx.y;               // E tile (4 tiles of 16)
  const int bh = blockIdx.z;
  const int b = bh >> 3, h = bh & 7;

  // Stage V[:, et*16 .. et*16+15] (1024 x 16 fp32 = 64 KB) transposed in LDS.
#pragma unroll
  for (int i = 0; i < 16; ++i) {
    const int j = tid + 256 * i;           // 0..4095
    const int s = j >> 2;
    const int e4 = (j & 3) * 4;
    const v4f t = *(const v4f*)&vv[(((size_t)b * SS + s) * HH + h) * EE + et * 16 + e4];
    sv[(e4 + 0) * 1028 + s] = t[0];
    sv[(e4 + 1) * 1028 + s] = t[1];
    sv[(e4 + 2) * 1028 + s] = t[2];
    sv[(e4 + 3) * 1028 + s] = t[3];
  }
  __syncthreads();

  const int wave = tid >> 5;
  const int lane = tid & 31;
  const int li = blockIdx.x * 8 + wave;    // query tile
  const int hi = lane >> 4;
  const int n  = lane & 15;
  const int mrow = li * 16 + n;
  const size_t arow = (((size_t)bh) * LL + mrow) * SS;
  const int svrow = n * 1028;

  v8f c = {};
  const int kend = li * 16 + 16;           // causal clip; multiple of 16
  for (int k0 = 0; k0 < kend; k0 += 16) {
#pragma unroll
    for (int kk4 = 0; kk4 < 16; kk4 += 4) {
      const v2f a  = *(const v2f*)&A[arow + k0 + kk4 + 2 * hi];
      const v2f bv = *(const v2f*)&sv[svrow + k0 + kk4 + 2 * hi];       // ds_load_b64
      c = __builtin_amdgcn_wmma_f32_16x16x4_f32(false, a, false, bv, (short)0, c,
                                                false, false);
    }
  }
#pragma unroll
  for (int r = 0; r < 8; ++r) {
    const int row = li * 16 + r + hi * 8;
    out[(((size_t)b * LL + row) * HH + h) * EE + et * 16 + n] = c[r];
  }
}

extern "C" void kernel_launch(void* const* d_in, const int* in_sizes, int n_in,
                              void* d_out, int out_size, void* d_ws, size_t ws_size,
                              hipStream_t stream) {
  (void)in_sizes; (void)n_in; (void)out_size; (void)ws_size;
  const float* q   = (const float*)d_in[0];  // [B,L,H,E]
  const float* k   = (const float*)d_in[1];  // [B,L,H,E]
  const float* v   = (const float*)d_in[2];  // [B,L,H,E]
  const float* mk  = (const float*)d_in[3];  // [H,1,3,3]
  const float* mk2 = (const float*)d_in[4];  // [H,1,3,3]
  const float* hk  = (const float*)d_in[5];  // [G,2,2]
  const float* wp  = (const float*)d_in[6];  // [H,H]
  float* out = (float*)d_out;                // [B,L,H,E]

  float* ws0 = (float*)d_ws;                           // scores / A_final
  float* ws1 = ws0 + (size_t)BB * HH * LL * SS;        // A (post-softmax)

  qk_wmma<<<dim3(LL / 16, (LL / 16) / 8, BB * HH), 256, 0, stream>>>(q, k, ws0);
  mix_softmax<<<dim3(LL, BB), 256, 0, stream>>>(ws0, mk, wp, ws1);
  conv2_headmix<<<dim3(LL, GG, BB), 256, 0, stream>>>(ws1, mk2, hk, ws0);
  av_wmma<<<dim3((LL / 16) / 8, EE / 16, BB * HH), 256, 0, stream>>>(ws0, v, out);
}